// FullExpertiseModel_6373731467819
// MI455X (gfx1250) — compile-verified
//
#include <hip/hip_runtime.h>

typedef __attribute__((ext_vector_type(2))) float v2f;
typedef __attribute__((ext_vector_type(8))) float v8f;

#define B_ROWS 16384
#define DIM    4096
#define NA     21

// GEMM kernel: out[b, a] = x[b, :] . W[a, :] + bias[a]
// One wave handles 32 rows (two 16-row M-tiles) x 32 padded cols (two N-tiles)
// over the full K=4096, using V_WMMA_F32_16X16X4_F32.
//
// WMMA f32 16x16x4 per-lane layout (wave32):
//   A (16x4):  M = lane%16, K = 2*(lane>>4) + vgpr  -> contiguous float2 per lane
//   B (4x16):  N = lane%16, K = 2*(lane>>4) + vgpr  -> contiguous float2 per lane
//   C/D (16x16): M = r + 8*(lane>>4), N = lane%16 for accumulator VGPR r
//
// Memory policy: x (268 MB) is streamed once -> non-temporal loads so it does
// not evict the L2-resident W (344 KB), which every wave re-reads.
__global__ __launch_bounds__(128) void fused_heads_gemm(
    const float* __restrict__ x, const float* __restrict__ W,
    const float* __restrict__ bias, float* __restrict__ out)
{
    const int lane = threadIdx.x & 31;
    const int wave = threadIdx.x >> 5;
    const int h    = lane >> 4;     // lane half: selects K base 2*h
    const int ln   = lane & 15;     // row (A) / col (B) within tile

    const long rowBase = (long)blockIdx.x * 128 + (long)wave * 32;

    // A-operand row pointers for the two M-tiles (rows are 16KB-aligned)
    const float* xr0 = x + (rowBase      + ln) * DIM + 2 * h;
    const float* xr1 = x + (rowBase + 16 + ln) * DIM + 2 * h;

    // B-operand row pointers: B[k,n] = W[n,k]; clamp cols 21..31 to row 20
    // (valid memory; those output columns are never stored).
    const int n0  = ln;
    const int n1  = 16 + ln;
    const int n1c = (n1 > NA - 1) ? (NA - 1) : n1;
    const float* wr0 = W + (long)n0  * DIM + 2 * h;
    const float* wr1 = W + (long)n1c * DIM + 2 * h;

    v8f acc00 = {}; // rows tile0 x cols tile0
    v8f acc01 = {}; // rows tile0 x cols tile1
    v8f acc10 = {}; // rows tile1 x cols tile0
    v8f acc11 = {}; // rows tile1 x cols tile1

#pragma unroll 8
    for (int k = 0; k < DIM; k += 4) {
        // Streaming operand: non-temporal (TH=NT) to avoid L2 pollution.
        v2f a0 = __builtin_nontemporal_load((const v2f*)(xr0 + k));
        v2f a1 = __builtin_nontemporal_load((const v2f*)(xr1 + k));
        // Resident operand: regular temporal so it stays hot in WGP$/L2.
        v2f b0 = *(const v2f*)(wr0 + k);
        v2f b1 = *(const v2f*)(wr1 + k);
        acc00 = __builtin_amdgcn_wmma_f32_16x16x4_f32(false, a0, false, b0,
                                                      (short)0, acc00, false, false);
        acc01 = __builtin_amdgcn_wmma_f32_16x16x4_f32(false, a0, false, b1,
                                                      (short)0, acc01, false, false);
        acc10 = __builtin_amdgcn_wmma_f32_16x16x4_f32(false, a1, false, b0,
                                                      (short)0, acc10, false, false);
        acc11 = __builtin_amdgcn_wmma_f32_16x16x4_f32(false, a1, false, b1,
                                                      (short)0, acc11, false, false);
    }

    // Epilogue: add bias, scatter-store valid columns of out[B, 21].
    {
        const float bias0 = bias[n0];
        const long  m0 = rowBase + 8 * h;        // M-tile 0
        const long  m1 = rowBase + 16 + 8 * h;   // M-tile 1
#pragma unroll
        for (int r = 0; r < 8; ++r)
            out[(m0 + r) * NA + n0] = acc00[r] + bias0;
#pragma unroll
        for (int r = 0; r < 8; ++r)
            out[(m1 + r) * NA + n0] = acc10[r] + bias0;
    }
    if (n1 < NA) {
        const float bias1 = bias[n1];
        const long  m0 = rowBase + 8 * h;
        const long  m1 = rowBase + 16 + 8 * h;
#pragma unroll
        for (int r = 0; r < 8; ++r)
            out[(m0 + r) * NA + n1] = acc01[r] + bias1;
#pragma unroll
        for (int r = 0; r < 8; ++r)
            out[(m1 + r) * NA + n1] = acc11[r] + bias1;
    }
}

// result[b] = sum_a out[b,a] * v[b,a] — tiny (2.75 MB) second pass; out is
// still hot in L2 from the GEMM epilogue.
__global__ __launch_bounds__(256) void weighted_sum(
    const float* __restrict__ out, const float* __restrict__ v,
    float* __restrict__ result)
{
    const long row = (long)blockIdx.x * 256 + threadIdx.x;
    const float* o  = out + row * NA;
    const float* vv = v   + row * NA;
    float s = 0.0f;
#pragma unroll
    for (int a = 0; a < NA; ++a)
        s = fmaf(o[a], vv[a], s);
    result[row] = s;
}

extern "C" void kernel_launch(void* const* d_in, const int* in_sizes, int n_in,
                              void* d_out, int out_size, void* d_ws, size_t ws_size,
                              hipStream_t stream) {
    // setup_inputs order: x [B,D], v [B,A], W [A,D], b [A]  (all float32)
    const float* x = (const float*)d_in[0];
    const float* v = (const float*)d_in[1];
    const float* W = (const float*)d_in[2];
    const float* b = (const float*)d_in[3];

    // Output tuple order: result [B] first, then out [B, A].
    float* result = (float*)d_out;
    float* out    = result + B_ROWS;

    // 128 rows per block (4 waves x 32 rows) -> 128 blocks, 512 waves.
    fused_heads_gemm<<<dim3(B_ROWS / 128), dim3(128), 0, stream>>>(x, W, b, out);
    // 16384 rows -> 64 blocks of 256 threads.
    weighted_sum<<<dim3(B_ROWS / 256), dim3(256), 0, stream>>>(out, v, result);
}